// SalienceAttention_88656714925488
// MI455X (gfx1250) — compile-verified
//
#include <hip/hip_runtime.h>
#include <math.h>

#define B_   32
#define C_   384
#define H3_  1152
#define NH_  12
#define HD_  32

typedef float v2f __attribute__((ext_vector_type(2)));
typedef float v8f __attribute__((ext_vector_type(8)));

__device__ __forceinline__ v8f wmma_f32(v2f a, v2f b, v8f c) {
  // D = A(16x4) * B(4x16) + C, fp32, wave32
  return __builtin_amdgcn_wmma_f32_16x16x4_f32(false, a, false, b, (short)0, c,
                                               false, false);
}

// Low 32 bits of a generic pointer to __shared__ == LDS byte offset.
__device__ __forceinline__ unsigned lds_off_u32(const void* p) {
  return (unsigned)(unsigned long long)p;
}

// CDNA5 async DMA: 16 contiguous bytes global -> LDS, tracked by ASYNCcnt.
__device__ __forceinline__ void async_g2l_b128(void* lds, const void* g) {
  asm volatile("global_load_async_to_lds_b128 %0, %1, off"
               :: "v"(lds_off_u32(lds)), "v"(g)
               : "memory");
}
__device__ __forceinline__ void wait_async0() {
  asm volatile("s_wait_asynccnt 0x0" ::: "memory");
}

// A-style fragment from a K-major (row-major [rows][K]) LDS buffer.
// lanes 0-15: k0,k0+1 ; lanes 16-31: k0+2,k0+3 ; row = row0 + (lane&15)
__device__ __forceinline__ v2f frag_kmajor(const float* s, int row0, int k0,
                                           int ld, int lane) {
  int half = lane >> 4, lr = lane & 15;
  const float* p = s + (row0 + lr) * ld + k0 + 2 * half;
  v2f f; f.x = p[0]; f.y = p[1]; return f;
}

// B fragment from a k-pair-interleaved buffer: element (k,c) stored at
// [(k>>1)*ldp2 + 2c + (k&1)] -> the (k,k+1) pair is one contiguous b64 load.
__device__ __forceinline__ v2f frag_paired(const float* s, int k0, int col0,
                                           int ldp2, int lane) {
  int half = lane >> 4, lr = lane & 15;
  const float* p = s + ((k0 >> 1) + half) * ldp2 + 2 * (col0 + lr);
  v2f f; f.x = p[0]; f.y = p[1]; return f;
}

// ---------------------------------------------------------------------------
// GEMM (NN): out[b][m][p] = act( sum_k W[m][k]*F[b][k][p] + bias[m] (+ add) )
// mode 0: bias only, 1: bias+exact GELU, 2: bias + add[b][m][p]
// grid: (Np/128, M/64, B), block 256 (8 waves, 2x4 wave grid, 32x32 per wave)
// ---------------------------------------------------------------------------
__global__ __launch_bounds__(256) void gemm_nn_kernel(
    const float* __restrict__ W, const float* __restrict__ F,
    const float* __restrict__ bias, const float* add, float* out,
    int M, int K, int Np, int mode) {
  __shared__ __align__(16) float sW[64 * 36];    // K-major weight tile
  __shared__ __align__(16) float sF[16 * 288];   // k-pair-interleaved act tile
  int tid = threadIdx.x;
  int lane = tid & 31, w = tid >> 5;
  int wm = w >> 2, wn = w & 3;
  int np0 = blockIdx.x * 128;
  int m0 = blockIdx.y * 64;
  int b = blockIdx.z;
  const float* Fb = F + (size_t)b * K * Np;

  v8f acc00 = {}, acc01 = {}, acc10 = {}, acc11 = {};
  for (int k0 = 0; k0 < K; k0 += 32) {
    // 64x32 weight tile via async DMA (rows are K-contiguous, 16B aligned)
#pragma unroll
    for (int i = 0; i < 2; ++i) {
      int e = tid + i * 256;           // 512 x 16B chunks
      int r = e >> 3, c4 = (e & 7) * 4;
      async_g2l_b128(&sW[r * 36 + c4], &W[(size_t)(m0 + r) * K + k0 + c4]);
    }
    // 32x128 activation tile -> paired layout (coalesced b32 in, b64 out)
#pragma unroll
    for (int i = 0; i < 4; ++i) {
      int u = tid + i * 256;           // 1024 columns-of-4k
      int q = u >> 7, c = u & 127;
      const float* gp = &Fb[(size_t)(k0 + 4 * q) * Np + np0 + c];
      float a0 = gp[0];
      float a1 = gp[(size_t)Np];
      float a2 = gp[2 * (size_t)Np];
      float a3 = gp[3 * (size_t)Np];
      float* d0 = &sF[(2 * q + 0) * 288 + 2 * c];
      float* d1 = &sF[(2 * q + 1) * 288 + 2 * c];
      d0[0] = a0; d0[1] = a1;
      d1[0] = a2; d1[1] = a3;
    }
    wait_async0();
    __syncthreads();
#pragma unroll
    for (int kk = 0; kk < 32; kk += 4) {
      v2f a0 = frag_kmajor(sW, wm * 32 + 0, kk, 36, lane);
      v2f a1 = frag_kmajor(sW, wm * 32 + 16, kk, 36, lane);
      v2f b0 = frag_paired(sF, kk, wn * 32 + 0, 288, lane);
      v2f b1 = frag_paired(sF, kk, wn * 32 + 16, 288, lane);
      acc00 = wmma_f32(a0, b0, acc00);
      acc01 = wmma_f32(a0, b1, acc01);
      acc10 = wmma_f32(a1, b0, acc10);
      acc11 = wmma_f32(a1, b1, acc11);
    }
    __syncthreads();
  }
  int half = lane >> 4, lr = lane & 15;
  float* ob = out + (size_t)b * M * Np;
  const float* ab = add ? add + (size_t)b * M * Np : nullptr;
#pragma unroll
  for (int mi = 0; mi < 2; ++mi)
#pragma unroll
    for (int ni = 0; ni < 2; ++ni) {
      v8f acc = (mi == 0) ? (ni == 0 ? acc00 : acc01)
                          : (ni == 0 ? acc10 : acc11);
      int row_base = m0 + wm * 32 + mi * 16 + 8 * half;
      int col = np0 + wn * 32 + ni * 16 + lr;
#pragma unroll
      for (int r = 0; r < 8; ++r) {
        int row = row_base + r;
        float v = acc[r] + bias[row];
        if (mode == 1) v = 0.5f * v * (1.0f + erff(v * 0.70710678118654752f));
        else if (mode == 2) v += ab[(size_t)row * Np + col];
        ob[(size_t)row * Np + col] = v;
      }
    }
}

// ---------------------------------------------------------------------------
// GEMM (NT): out[m][n] = sum_k A[m][k] * Wt[n][k] + bias[n]
// grid: (Ncols/128, ceil(Mrows/64)), block 256
// ---------------------------------------------------------------------------
__global__ __launch_bounds__(256) void gemm_nt_kernel(
    const float* __restrict__ A, const float* __restrict__ Wt,
    const float* __restrict__ bias, float* __restrict__ out,
    int Mrows, int K, int Ncols) {
  __shared__ __align__(16) float sA[64 * 36];
  __shared__ __align__(16) float sW[128 * 36];
  int tid = threadIdx.x, lane = tid & 31, w = tid >> 5;
  int wm = w >> 2, wn = w & 3;
  int n0 = blockIdx.x * 128;
  int m0 = blockIdx.y * 64;
  v8f acc00 = {}, acc01 = {}, acc10 = {}, acc11 = {};
  for (int k0 = 0; k0 < K; k0 += 32) {
    // A tile: clamp out-of-range rows (padded output rows are never stored)
#pragma unroll
    for (int i = 0; i < 2; ++i) {
      int e = tid + i * 256;
      int r = e >> 3, c4 = (e & 7) * 4;
      int m = m0 + r; if (m >= Mrows) m = Mrows - 1;
      async_g2l_b128(&sA[r * 36 + c4], &A[(size_t)m * K + k0 + c4]);
    }
#pragma unroll
    for (int i = 0; i < 4; ++i) {
      int e = tid + i * 256;
      int r = e >> 3, c4 = (e & 7) * 4;
      async_g2l_b128(&sW[r * 36 + c4], &Wt[(size_t)(n0 + r) * K + k0 + c4]);
    }
    wait_async0();
    __syncthreads();
#pragma unroll
    for (int kk = 0; kk < 32; kk += 4) {
      v2f a0 = frag_kmajor(sA, wm * 32 + 0, kk, 36, lane);
      v2f a1 = frag_kmajor(sA, wm * 32 + 16, kk, 36, lane);
      v2f b0 = frag_kmajor(sW, wn * 32 + 0, kk, 36, lane);
      v2f b1 = frag_kmajor(sW, wn * 32 + 16, kk, 36, lane);
      acc00 = wmma_f32(a0, b0, acc00);
      acc01 = wmma_f32(a0, b1, acc01);
      acc10 = wmma_f32(a1, b0, acc10);
      acc11 = wmma_f32(a1, b1, acc11);
    }
    __syncthreads();
  }
  int half = lane >> 4, lr = lane & 15;
#pragma unroll
  for (int mi = 0; mi < 2; ++mi)
#pragma unroll
    for (int ni = 0; ni < 2; ++ni) {
      v8f acc = (mi == 0) ? (ni == 0 ? acc00 : acc01)
                          : (ni == 0 ? acc10 : acc11);
      int row_base = m0 + wm * 32 + mi * 16 + 8 * half;
      int col = n0 + wn * 32 + ni * 16 + lr;
#pragma unroll
      for (int r = 0; r < 8; ++r) {
        int row = row_base + r;
        if (row < Mrows) out[(size_t)row * Ncols + col] = acc[r] + bias[col];
      }
    }
}

// ---------------------------------------------------------------------------
// Fused multi-head attention for one (batch, head). N tokens, head dim 32.
// grid (NH_, B_), block 128 (4 waves; each wave owns whole 16-row q-tiles)
// ---------------------------------------------------------------------------
template <int N, int NPAD>
__global__ __launch_bounds__(128) void attn_kernel(const float* __restrict__ qkv,
                                                   float* __restrict__ outp) {
  constexpr int NT = NPAD / 16;
  __shared__ __align__(16) float sK[NPAD * 32];          // K-major keys
  __shared__ __align__(16) float sV[(NPAD / 2) * 96];    // paired values
  __shared__ __align__(16) float sP[4 * 16 * NPAD];      // per-wave P strip
  int tid = threadIdx.x, lane = tid & 31, w = tid >> 5;
  int h = blockIdx.x, b = blockIdx.y;
  const float* base = qkv + (size_t)b * N * H3_;
  for (int e = tid; e < NPAD * 32; e += 128) {
    int n = e >> 5, d = e & 31;
    float kv = 0.f, vv = 0.f;
    if (n < N) {
      const float* rp = base + (size_t)n * H3_ + h * HD_ + d;
      kv = rp[C_];
      vv = rp[2 * C_];
    }
    sK[n * 32 + d] = kv;
    sV[(n >> 1) * 96 + 2 * d + (n & 1)] = vv;
  }
  __syncthreads();
  int half = lane >> 4, lr = lane & 15;
  const float scale = 0.17677669529663687f;  // 1/sqrt(32)
  const v8f zero8 = {};
  for (int qt = w; qt < NT; qt += 4) {
    int m0 = qt * 16;
    v8f sacc[NT];
#pragma unroll
    for (int j = 0; j < NT; ++j) sacc[j] = zero8;
    // S = Q * K^T (q fragments straight from global, keys from LDS)
    int qrow = m0 + lr; if (qrow >= N) qrow = N - 1;
#pragma unroll
    for (int kk = 0; kk < HD_; kk += 4) {
      const float* qp = base + (size_t)qrow * H3_ + h * HD_ + kk + 2 * half;
      v2f a; a.x = qp[0]; a.y = qp[1];
#pragma unroll
      for (int j = 0; j < NT; ++j) {
        v2f bf = frag_kmajor(sK, j * 16, kk, 32, lane);
        sacc[j] = wmma_f32(a, bf, sacc[j]);
      }
    }
    // row softmax in the C/D lane layout (16-lane groups reduce across keys)
#pragma unroll
    for (int r = 0; r < 8; ++r) {
      float vals[NT];
      float rmax = -INFINITY;
#pragma unroll
      for (int j = 0; j < NT; ++j) {
        int n = j * 16 + lr;
        float v = (n < N) ? sacc[j][r] * scale : -INFINITY;
        vals[j] = v;
        rmax = fmaxf(rmax, v);
      }
#pragma unroll
      for (int m = 1; m < 16; m <<= 1) rmax = fmaxf(rmax, __shfl_xor(rmax, m));
      float rsum = 0.f;
#pragma unroll
      for (int j = 0; j < NT; ++j) {
        float e = (vals[j] == -INFINITY) ? 0.f : __expf(vals[j] - rmax);
        vals[j] = e;
        rsum += e;
      }
#pragma unroll
      for (int m = 1; m < 16; m <<= 1) rsum += __shfl_xor(rsum, m);
      float inv = 1.0f / rsum;
      int mlocal = r + 8 * half;
#pragma unroll
      for (int j = 0; j < NT; ++j)
        sP[(w * 16 + mlocal) * NPAD + j * 16 + lr] = vals[j] * inv;
    }
    // O = P * V  (wave-private sP strip; same-wave LDS ops are ordered)
    v8f o0 = zero8, o1 = zero8;
#pragma unroll
    for (int kk = 0; kk < NPAD; kk += 4) {
      v2f a = frag_kmajor(sP + w * 16 * NPAD, 0, kk, NPAD, lane);
      v2f b0 = frag_paired(sV, kk, 0, 96, lane);
      v2f b1 = frag_paired(sV, kk, 16, 96, lane);
      o0 = wmma_f32(a, b0, o0);
      o1 = wmma_f32(a, b1, o1);
    }
#pragma unroll
    for (int r = 0; r < 8; ++r) {
      int row = m0 + r + 8 * half;
      if (row < N) {
        float* op = outp + (size_t)(b * N + row) * C_ + h * HD_;
        op[lr] = o0[r];
        op[16 + lr] = o1[r];
      }
    }
  }
}

// ---------------------------------------------------------------------------
// salience scores + per-channel sums
// ---------------------------------------------------------------------------
__global__ void scores_colsum_kernel(const float* __restrict__ feat,
                                     float* __restrict__ scores,
                                     float* __restrict__ colsum, int Np) {
  int b = blockIdx.x, tid = threadIdx.x;
  const float* fb = feat + (size_t)b * C_ * Np;
  for (int p = tid; p < Np; p += 256) {
    float s = 0.f;
    for (int c = 0; c < C_; ++c) {
      float v = fb[(size_t)c * Np + p];
      s += v * v;
    }
    scores[b * Np + p] = s;
  }
  for (int c = tid; c < C_; c += 256) {
    const float* row = fb + (size_t)c * Np;
    float s = 0.f;
    for (int p = 0; p < Np; ++p) s += row[p];
    colsum[b * C_ + c] = s;
  }
}

// stable top-k (jax.lax.top_k semantics: descending, ties -> lower index)
__global__ void topk_kernel(const float* __restrict__ scores,
                            int* __restrict__ topidx, int Np, int k) {
  __shared__ float ls[1024];
  int b = blockIdx.x, tid = threadIdx.x;
  for (int p = tid; p < Np; p += 256) ls[p] = scores[b * Np + p];
  __syncthreads();
  for (int p = tid; p < Np; p += 256) {
    float s = ls[p];
    int rank = 0;
    for (int q = 0; q < Np; ++q) {
      float sq = ls[q];
      rank += (sq > s) || (sq == s && q < p);
    }
    if (rank < k) topidx[b * k + rank] = p;
  }
}

// tokens[b][n][c]: gathered top tokens + background mean row
__global__ void gather_kernel(const float* __restrict__ feat,
                              const int* __restrict__ topidx,
                              const float* __restrict__ colsum,
                              float* __restrict__ tokens, int Np, int k) {
  __shared__ int lidx[128];
  int b = blockIdx.x, tid = threadIdx.x;
  if (tid < k) lidx[tid] = topidx[b * k + tid];
  __syncthreads();
  int N = k + 1;
  float invden = 1.0f / (float)(Np - k);
  for (int c = tid; c < C_; c += 256) {
    const float* row = feat + ((size_t)b * C_ + c) * Np;
    float acc = 0.f;
    for (int n = 0; n < k; ++n) {
      float v = row[lidx[n]];
      tokens[(size_t)(b * N + n) * C_ + c] = v;
      acc += v;
    }
    tokens[(size_t)(b * N + k) * C_ + c] = (colsum[b * C_ + c] - acc) * invden;
  }
}

// fill with background token, scatter selected tokens back
__global__ void restore_kernel(const float* __restrict__ tokens_out,
                               const int* __restrict__ topidx,
                               float* __restrict__ t, int Np, int k) {
  int bc = blockIdx.x;
  int b = bc / C_, c = bc % C_;
  int N = k + 1;
  int tid = threadIdx.x;
  float bg = tokens_out[(size_t)(b * N + k) * C_ + c];
  float* trow = t + (size_t)bc * Np;
  for (int p = tid; p < Np; p += 256) trow[p] = bg;
  __syncthreads();
  for (int n = tid; n < k; n += 256)
    trow[topidx[b * k + n]] = tokens_out[(size_t)(b * N + n) * C_ + c];
}

// depthwise 3x3 SAME conv + bias
__global__ void dwconv_kernel(const float* __restrict__ y1,
                              const float* __restrict__ cw,
                              const float* __restrict__ cb,
                              float* __restrict__ ydw, int H, int W) {
  int bc = blockIdx.x;
  int c = bc % C_;
  int Np = H * W;
  const float* in = y1 + (size_t)bc * Np;
  float* out = ydw + (size_t)bc * Np;
  float wv[9];
#pragma unroll
  for (int i = 0; i < 9; ++i) wv[i] = cw[c * 9 + i];
  float bias = cb[c];
  for (int p = threadIdx.x; p < Np; p += 256) {
    int y = p / W, x = p % W;
    float s = bias;
#pragma unroll
    for (int dy = -1; dy <= 1; ++dy)
#pragma unroll
      for (int dx = -1; dx <= 1; ++dx) {
        int yy = y + dy, xx = x + dx;
        if (yy >= 0 && yy < H && xx >= 0 && xx < W)
          s += wv[(dy + 1) * 3 + (dx + 1)] * in[yy * W + xx];
      }
    out[p] = s;
  }
}

// fold BatchNorm into fc1:  W1eff = fc1_w * scale,  b1eff = fc1_b + fc1_w @ shift
__global__ void prep_w1_kernel(const float* __restrict__ fc1_w,
                               const float* __restrict__ fc1_b,
                               const float* __restrict__ gamma,
                               const float* __restrict__ beta,
                               const float* __restrict__ mean,
                               const float* __restrict__ var,
                               float* __restrict__ w1eff,
                               float* __restrict__ b1eff) {
  __shared__ float red[128];
  int o = blockIdx.x, tid = threadIdx.x;
  float part = 0.f;
  for (int c = tid; c < C_; c += 128) {
    float sc = gamma[c] * rsqrtf(var[c] + 1e-5f);
    float sh = beta[c] - mean[c] * sc;
    float w = fc1_w[o * C_ + c];
    w1eff[o * C_ + c] = w * sc;
    part += w * sh;
  }
  red[tid] = part;
  __syncthreads();
  for (int s = 64; s > 0; s >>= 1) {
    if (tid < s) red[tid] += red[tid + s];
    __syncthreads();
  }
  if (tid == 0) b1eff[o] = fc1_b[o] + red[0];
}

// ---------------------------------------------------------------------------
extern "C" void kernel_launch(void* const* d_in, const int* in_sizes, int n_in,
                              void* d_out, int out_size, void* d_ws,
                              size_t ws_size, hipStream_t stream) {
  (void)in_sizes; (void)n_in; (void)out_size; (void)ws_size;
  const float* z = (const float*)d_in[0];
  const float* x = (const float*)d_in[1];
  const float* qkv_w = (const float*)d_in[2];
  const float* qkv_b = (const float*)d_in[3];
  const float* po_w = (const float*)d_in[4];
  const float* po_b = (const float*)d_in[5];
  const float* proj_w = (const float*)d_in[6];
  const float* proj_b = (const float*)d_in[7];
  const float* conv_w = (const float*)d_in[8];
  const float* conv_b = (const float*)d_in[9];
  const float* fc1_w = (const float*)d_in[10];
  const float* fc1_b = (const float*)d_in[11];
  const float* fc2_w = (const float*)d_in[12];
  const float* fc2_b = (const float*)d_in[13];
  const float* bn_g = (const float*)d_in[14];
  const float* bn_b = (const float*)d_in[15];
  const float* bn_m = (const float*)d_in[16];
  const float* bn_v = (const float*)d_in[17];
  // k_template=32, k_search=98 (python scalars; fixed by reference constants)

  char* p = (char*)d_ws;
  auto alloc = [&](size_t bytes) {
    char* r = p;
    p += (bytes + 255) & ~(size_t)255;
    return r;
  };
  float* w1eff = (float*)alloc((size_t)C_ * C_ * 4);
  float* b1eff = (float*)alloc((size_t)C_ * 4);
  float* scores = (float*)alloc((size_t)B_ * 1024 * 4);
  float* colsum = (float*)alloc((size_t)B_ * C_ * 4);
  int* topidx = (int*)alloc((size_t)B_ * 98 * 4);
  float* tokens = (float*)alloc((size_t)B_ * 99 * C_ * 4);
  float* qkvb = (float*)alloc((size_t)B_ * 99 * H3_ * 4);
  float* attnout = (float*)alloc((size_t)B_ * 99 * C_ * 4);
  float* tokout = (float*)alloc((size_t)B_ * 99 * C_ * 4);
  float* tbuf = (float*)alloc((size_t)B_ * C_ * 1024 * 4);
  float* y1 = (float*)alloc((size_t)B_ * C_ * 1024 * 4);
  float* ydw = (float*)alloc((size_t)B_ * C_ * 1024 * 4);

  prep_w1_kernel<<<C_, 128, 0, stream>>>(fc1_w, fc1_b, bn_g, bn_b, bn_m, bn_v,
                                         w1eff, b1eff);

  float* outf = (float*)d_out;
  float* z_out = outf;
  float* x_out = outf + (size_t)B_ * C_ * 256;

  struct Branch { const float* feat; int Np, H, k, N; float* out; };
  Branch br[2] = {{z, 256, 16, 32, 33, z_out}, {x, 1024, 32, 98, 99, x_out}};

  for (int i = 0; i < 2; ++i) {
    const float* feat = br[i].feat;
    int Np = br[i].Np, H = br[i].H, k = br[i].k, N = br[i].N;
    float* outb = br[i].out;
    int BN = B_ * N;
    int mt = (BN + 63) / 64;

    scores_colsum_kernel<<<B_, 256, 0, stream>>>(feat, scores, colsum, Np);
    topk_kernel<<<B_, 256, 0, stream>>>(scores, topidx, Np, k);
    gather_kernel<<<B_, 256, 0, stream>>>(feat, topidx, colsum, tokens, Np, k);

    // qkv = tokens @ qkv_w^T + qkv_b   [(B*N) x 1152]
    gemm_nt_kernel<<<dim3(H3_ / 128, mt), 256, 0, stream>>>(
        tokens, qkv_w, qkv_b, qkvb, BN, C_, H3_);

    if (N == 33)
      attn_kernel<33, 48><<<dim3(NH_, B_), 128, 0, stream>>>(qkvb, attnout);
    else
      attn_kernel<99, 112><<<dim3(NH_, B_), 128, 0, stream>>>(qkvb, attnout);

    // tokens_out = attnout @ proj_out_w^T + proj_out_b
    gemm_nt_kernel<<<dim3(C_ / 128, mt), 256, 0, stream>>>(
        attnout, po_w, po_b, tokout, BN, C_, C_);

    restore_kernel<<<B_ * C_, 256, 0, stream>>>(tokout, topidx, tbuf, Np, k);

    // y1 = gelu(W1eff @ feat + b1eff)   (BN folded into fc1)
    gemm_nn_kernel<<<dim3(Np / 128, C_ / 64, B_), 256, 0, stream>>>(
        w1eff, feat, b1eff, nullptr, y1, C_, C_, Np, 1);

    dwconv_kernel<<<B_ * C_, 256, 0, stream>>>(y1, conv_w, conv_b, ydw, H, H);

    // tbuf = fc2_w @ ydw + fc2_b + tbuf  (restored + conv branch)
    gemm_nn_kernel<<<dim3(Np / 128, C_ / 64, B_), 256, 0, stream>>>(
        fc2_w, ydw, fc2_b, tbuf, tbuf, C_, C_, Np, 2);

    // out = proj_w @ tbuf + proj_b
    gemm_nn_kernel<<<dim3(Np / 128, C_ / 64, B_), 256, 0, stream>>>(
        proj_w, tbuf, proj_b, nullptr, outb, C_, C_, Np, 0);
  }
}